// Flen_16114717295174
// MI455X (gfx1250) — compile-verified
//
#include <hip/hip_runtime.h>
#include <hip/hip_bf16.h>

// ---------------------------------------------------------------------------
// Graph-transformer encoder for MI455X (gfx1250), fp32 WMMA (16x16x4) path.
// Register-blocked fragment reuse + vectorized LDS staging + weight prefetch.
// ---------------------------------------------------------------------------

typedef float v2f __attribute__((ext_vector_type(2)));
typedef float v4f __attribute__((ext_vector_type(4)));
typedef float v8f __attribute__((ext_vector_type(8)));

#define BQ   128   // batch
#define SQ   128   // seq
#define DQ   256   // embed
#define HQ   8     // heads
#define HDQ  32    // head dim
#define FFNQ 2048  // ffn dim
#define BSQ  (BQ * SQ)

#define QLD  36    // LDS stride for [128][32] tiles (16B-aligned, conflict-free)
#define SSTR 132   // LDS stride for score rows (128 + 4)
#define CLD  36    // LDS stride for ctx staging
#define ALD  260   // LDS stride for 256-wide row tiles
#define HLD  68    // LDS stride for 64-wide H tiles

__device__ __forceinline__ v8f wmma4(v2f a, v2f b, v8f c) {
  // D = A(16x4 f32) * B(4x16 f32) + C(16x16 f32)
  return __builtin_amdgcn_wmma_f32_16x16x4_f32(
      /*neg_a=*/false, a, /*neg_b=*/false, b,
      /*c_mod=*/(short)0, c, /*reuse_a=*/false, /*reuse_b=*/false);
}

// A-fragment: element(m, kidx) = base[m*ld + kidx]  (row-major MxK)
__device__ __forceinline__ v2f frag_ak(const float* base, int ld, int m0, int kk, int lane) {
  int m  = m0 + (lane & 15);
  int ks = kk + ((lane < 16) ? 0 : 2);
  v2f a;
  a.x = base[(size_t)m * ld + ks];
  a.y = base[(size_t)m * ld + ks + 1];
  return a;
}

// B-fragment: element(kidx, n) = base[kidx*ld + n]  (row-major KxN)
__device__ __forceinline__ v2f frag_kb(const float* base, int ld, int kk, int n0, int lane) {
  int n  = n0 + (lane & 15);
  int ks = kk + ((lane < 16) ? 0 : 2);
  v2f b;
  b.x = base[(size_t)ks * ld + n];
  b.y = base[(size_t)(ks + 1) * ld + n];
  return b;
}

// LayerNorm over 16 rows x 256 cols staged in LDS, write to global.
__device__ __forceinline__ void ln_rows_write(const float* Ys, int ldy,
                                              const float* g, const float* beta,
                                              float* out, int row0) {
  const int tid = threadIdx.x, wave = tid >> 5, lane = tid & 31;
  for (int p = 0; p < 4; ++p) {
    int rl = wave * 4 + p;
    const float* yr = Ys + (size_t)rl * ldy;
    float s = 0.f;
    for (int c = lane; c < DQ; c += 32) s += yr[c];
    for (int off = 16; off; off >>= 1) s += __shfl_xor(s, off, 32);
    float mu = s * (1.f / DQ);
    float vv = 0.f;
    for (int c = lane; c < DQ; c += 32) { float d = yr[c] - mu; vv = fmaf(d, d, vv); }
    for (int off = 16; off; off >>= 1) vv += __shfl_xor(vv, off, 32);
    float rstd = rsqrtf(vv * (1.f / DQ) + 1e-5f);
    for (int c = lane; c < DQ; c += 32)
      out[(size_t)(row0 + rl) * DQ + c] = (yr[c] - mu) * rstd * g[c] + beta[c];
  }
}

// ---------------------------------------------------------------------------
// GEMM: C[M,N] = A[M,K] @ B[K,N] + bias[N]   (optional relu)
// grid = (M/64, N/64), block = 128.  Each wave: 4 row-tiles x 1 col-tile,
// B fragment reused across the 4 row-tiles.
// ---------------------------------------------------------------------------
__global__ __launch_bounds__(128) void gemm_kernel(
    const float* __restrict__ A, const float* __restrict__ Bw,
    const float* __restrict__ bias, float* __restrict__ C,
    int M, int N, int K, int relu) {
  extern __shared__ float smem[];
  float* As = smem;  // [64][ALD]
  const int tid = threadIdx.x, wave = tid >> 5, lane = tid & 31;
  const int row0 = blockIdx.x * 64;
  const int col0 = blockIdx.y * 64 + wave * 16;
  const int kq = K >> 2;  // K/4 float4s per row

  for (int idx = tid; idx < 64 * kq; idx += 128) {
    int r = idx / kq, c4 = idx % kq;
    *reinterpret_cast<v4f*>(&As[r * ALD + c4 * 4]) =
        *reinterpret_cast<const v4f*>(A + (size_t)(row0 + r) * K + c4 * 4);
  }
  __syncthreads();

  v8f acc[4] = {};
  for (int kk = 0; kk < K; kk += 4) {
    v2f b = frag_kb(Bw, N, kk, col0, lane);
    #pragma unroll
    for (int rt = 0; rt < 4; ++rt) {
      v2f a = frag_ak(As, ALD, rt * 16, kk, lane);
      acc[rt] = wmma4(a, b, acc[rt]);
    }
  }

  const int cc = col0 + (lane & 15);
  const float bb = bias ? bias[cc] : 0.f;
  #pragma unroll
  for (int rt = 0; rt < 4; ++rt) {
    int base = rt * 16 + ((lane < 16) ? 0 : 8);
    #pragma unroll
    for (int r = 0; r < 8; ++r) {
      float v = acc[rt][r] + bb;
      if (relu) v = fmaxf(v, 0.f);
      C[(size_t)(row0 + base + r) * N + cc] = v;
    }
  }
}

// ---------------------------------------------------------------------------
// Edge-biased attention, one block per (b,h).  128 threads.
// ---------------------------------------------------------------------------
__global__ __launch_bounds__(128) void attn_kernel(
    const float* __restrict__ qg, const float* __restrict__ kg, const float* __restrict__ vg,
    const float* __restrict__ ek, const float* __restrict__ ev,
    const unsigned char* __restrict__ mask, float* __restrict__ ctx) {
  extern __shared__ float smem[];
  float* qs  = smem;                  // [128][QLD]
  float* ks_ = qs + SQ * QLD;         // [128][QLD]
  float* vs  = ks_ + SQ * QLD;        // [128][QLD]
  float* ssc = vs + SQ * QLD;         // [32][SSTR]
  float* cxs = ssc + 32 * SSTR;       // [32][CLD]

  const int tid = threadIdx.x, wave = tid >> 5, lane = tid & 31;
  const int bh = blockIdx.x, b = bh >> 3, h = bh & 7;
  const float scale = 0.17677669529663687f;  // 1/sqrt(32)

  // stage q, k, v head slices into LDS (b128 vectorized)
  for (int idx = tid; idx < SQ * 8; idx += 128) {
    int i = idx >> 3, d4 = (idx & 7) * 4;
    size_t gp = (size_t)(b * SQ + i) * DQ + h * HDQ + d4;
    *reinterpret_cast<v4f*>(&qs[i * QLD + d4])  = *reinterpret_cast<const v4f*>(qg + gp);
    *reinterpret_cast<v4f*>(&ks_[i * QLD + d4]) = *reinterpret_cast<const v4f*>(kg + gp);
    *reinterpret_cast<v4f*>(&vs[i * QLD + d4])  = *reinterpret_cast<const v4f*>(vg + gp);
  }
  __syncthreads();

  for (int ib = 0; ib < 4; ++ib) {
    const int i0 = ib * 32;

    // (a) edge-key bias: bias[i][j] = sum_d ek[j,i,d] * q[j,d]
    for (int idx = tid; idx < 32 * SQ; idx += 128) {
      int il = idx >> 7, j = idx & 127;
      int i = i0 + il;
      const float* ekp = ek + ((size_t)j * SQ + i) * HDQ;
      const float* qj = qs + j * QLD;
      float acc = 0.f;
      for (int d = 0; d < HDQ; ++d) acc = fmaf(ekp[d], qj[d], acc);
      ssc[il * SSTR + j] = acc;
    }
    __syncthreads();

    // (b) scores = q @ k^T via WMMA; one Q fragment feeds 4 tiles per wave
    {
      const int i0t = (wave >> 1) * 16;       // local row offset (0 or 16)
      const int j0b = (wave & 1) * 64;        // 4 col-tiles starting here
      v8f acc[4] = {};
      for (int kk = 0; kk < HDQ; kk += 4) {
        v2f a = frag_ak(qs, QLD, i0 + i0t, kk, lane);
        #pragma unroll
        for (int n = 0; n < 4; ++n) {
          v2f bf = frag_ak(ks_, QLD, j0b + n * 16, kk, lane);  // B = k^T
          acc[n] = wmma4(a, bf, acc[n]);
        }
      }
      #pragma unroll
      for (int n = 0; n < 4; ++n) {
        int base = i0t + ((lane < 16) ? 0 : 8);
        int cc = j0b + n * 16 + (lane & 15);
        #pragma unroll
        for (int r = 0; r < 8; ++r) {
          int rl = base + r;
          int gi = i0 + rl;
          float bias = ssc[rl * SSTR + cc];
          float mb = mask[(size_t)bh * SQ * SQ + (size_t)gi * SQ + cc]
                         ? -__builtin_inff() : 0.f;
          ssc[rl * SSTR + cc] = fmaf(acc[n][r] + bias, scale, mb);
        }
      }
    }
    __syncthreads();

    // (c) softmax over j (each wave: 8 rows)
    for (int rl = wave; rl < 32; rl += 4) {
      float* row = ssc + rl * SSTR;
      float vals[4], m = -__builtin_inff();
      for (int c = 0; c < 4; ++c) { vals[c] = row[lane + c * 32]; m = fmaxf(m, vals[c]); }
      for (int off = 16; off; off >>= 1) m = fmaxf(m, __shfl_xor(m, off, 32));
      float s = 0.f;
      for (int c = 0; c < 4; ++c) { vals[c] = __expf(vals[c] - m); s += vals[c]; }
      for (int off = 16; off; off >>= 1) s += __shfl_xor(s, off, 32);
      float inv = 1.f / s;
      for (int c = 0; c < 4; ++c) row[lane + c * 32] = vals[c] * inv;
    }
    __syncthreads();

    // (d) ctx1 = attn @ v via WMMA (2x2 tiles, one per wave)
    {
      int i0t = (wave >> 1) * 16;
      int d0 = (wave & 1) * 16;
      v8f acc = {};
      for (int kk = 0; kk < SQ; kk += 4) {
        v2f a = frag_ak(ssc, SSTR, i0t, kk, lane);
        v2f bf = frag_kb(vs, QLD, kk, d0, lane);
        acc = wmma4(a, bf, acc);
      }
      int base = i0t + ((lane < 16) ? 0 : 8);
      int cc = d0 + (lane & 15);
      #pragma unroll
      for (int r = 0; r < 8; ++r) cxs[(base + r) * CLD + cc] = acc[r];
    }
    __syncthreads();

    // (e) ctx2 += attn @ edge_value (per-row GEMV) and write out
    for (int p = 0; p < 8; ++p) {
      int il = wave + p * 4;
      int i = i0 + il;
      float acc = cxs[il * CLD + lane];
      const float* evp = ev + (size_t)i * SQ * HDQ + lane;
      const float* arow = ssc + il * SSTR;
      for (int j = 0; j < SQ; ++j) acc = fmaf(arow[j], evp[(size_t)j * HDQ], acc);
      ctx[(size_t)(b * SQ + i) * DQ + h * HDQ + lane] = acc;
    }
    __syncthreads();
  }
}

// ---------------------------------------------------------------------------
// out = LayerNorm(ctx @ Wo + bo + residual).  grid = M/16, block = 128.
// One A fragment feeds 4 col-tiles per wave.
// ---------------------------------------------------------------------------
__global__ __launch_bounds__(128) void proj_ln_kernel(
    const float* __restrict__ Ain, const float* __restrict__ Bw,
    const float* __restrict__ bias, const float* __restrict__ res,
    const float* __restrict__ g, const float* __restrict__ beta,
    float* __restrict__ out) {
  extern __shared__ float smem[];
  float* As = smem;            // [16][ALD]
  float* Ys = As + 16 * ALD;   // [16][ALD]
  const int tid = threadIdx.x, wave = tid >> 5, lane = tid & 31;
  const int row0 = blockIdx.x * 16;

  for (int idx = tid; idx < 16 * 64; idx += 128) {
    int r = idx >> 6, c4 = (idx & 63) * 4;
    *reinterpret_cast<v4f*>(&As[r * ALD + c4]) =
        *reinterpret_cast<const v4f*>(Ain + (size_t)(row0 + r) * DQ + c4);
  }
  __syncthreads();

  v8f acc[4] = {};
  for (int kk = 0; kk < DQ; kk += 4) {
    v2f a = frag_ak(As, ALD, 0, kk, lane);
    #pragma unroll
    for (int t = 0; t < 4; ++t) {
      v2f bf = frag_kb(Bw, DQ, kk, wave * 64 + t * 16, lane);
      acc[t] = wmma4(a, bf, acc[t]);
    }
  }
  #pragma unroll
  for (int t = 0; t < 4; ++t) {
    int base = (lane < 16) ? 0 : 8;
    int cc = wave * 64 + t * 16 + (lane & 15);
    #pragma unroll
    for (int r = 0; r < 8; ++r) {
      int rl = base + r;
      Ys[rl * ALD + cc] = acc[t][r] + bias[cc] + res[(size_t)(row0 + rl) * DQ + cc];
    }
  }
  __syncthreads();
  ln_rows_write(Ys, ALD, g, beta, out, row0);
}

// ---------------------------------------------------------------------------
// Fused FFN: out = LN(2*(relu(X@W1+b1)@W2 + b2)).  grid = M/16, block = 128.
// ---------------------------------------------------------------------------
__global__ __launch_bounds__(128) void ffn_kernel(
    const float* __restrict__ X, const float* __restrict__ W1, const float* __restrict__ b1,
    const float* __restrict__ W2, const float* __restrict__ b2,
    const float* __restrict__ g, const float* __restrict__ beta,
    float* __restrict__ out) {
  extern __shared__ float smem[];
  float* Xs = smem;             // [16][ALD]
  float* Hs = Xs + 16 * ALD;    // [16][HLD]
  float* Ys = Hs + 16 * HLD;    // [16][ALD]
  const int tid = threadIdx.x, wave = tid >> 5, lane = tid & 31;
  const int row0 = blockIdx.x * 16;

  for (int idx = tid; idx < 16 * 64; idx += 128) {
    int r = idx >> 6, c4 = (idx & 63) * 4;
    *reinterpret_cast<v4f*>(&Xs[r * ALD + c4]) =
        *reinterpret_cast<const v4f*>(X + (size_t)(row0 + r) * DQ + c4);
  }
  __syncthreads();

  v8f yacc[4] = {};
  for (int c = 0; c < FFNQ / 64; ++c) {
    const int f0 = c * 64;
    // prefetch next chunk's weights into cache (global_prefetch_b8)
    if (c + 1 < FFNQ / 64) {
      __builtin_prefetch(W2 + (size_t)(f0 + 64) * DQ + (size_t)tid * 128, 0, 3);
      __builtin_prefetch(W1 + (size_t)(tid * 2) * FFNQ + f0 + 64, 0, 3);
      __builtin_prefetch(W1 + (size_t)(tid * 2 + 1) * FFNQ + f0 + 64, 0, 3);
    }
    // H tile: each wave computes 16 of the 64 chunk columns
    {
      int hc0 = wave * 16;
      v8f hacc = {};
      for (int kk = 0; kk < DQ; kk += 4) {
        v2f a = frag_ak(Xs, ALD, 0, kk, lane);
        v2f bf = frag_kb(W1, FFNQ, kk, f0 + hc0, lane);
        hacc = wmma4(a, bf, hacc);
      }
      int base = (lane < 16) ? 0 : 8;
      int cc = hc0 + (lane & 15);
      float bb = b1[f0 + cc];
      #pragma unroll
      for (int r = 0; r < 8; ++r)
        Hs[(base + r) * HLD + cc] = fmaxf(hacc[r] + bb, 0.f);
    }
    __syncthreads();
    // Y += H @ W2 chunk; one H fragment feeds 4 col-tiles per wave
    for (int kk = 0; kk < 64; kk += 4) {
      v2f a = frag_ak(Hs, HLD, 0, kk, lane);
      #pragma unroll
      for (int t = 0; t < 4; ++t) {
        v2f bf = frag_kb(W2, DQ, f0 + kk, wave * 64 + t * 16, lane);
        yacc[t] = wmma4(a, bf, yacc[t]);
      }
    }
    __syncthreads();
  }

  #pragma unroll
  for (int t = 0; t < 4; ++t) {
    int base = (lane < 16) ? 0 : 8;
    int cc = wave * 64 + t * 16 + (lane & 15);
    #pragma unroll
    for (int r = 0; r < 8; ++r)
      Ys[(base + r) * ALD + cc] = 2.f * (yacc[t][r] + b2[cc]);
  }
  __syncthreads();
  ln_rows_write(Ys, ALD, g, beta, out, row0);
}

// ---------------------------------------------------------------------------
// [B,S,D] -> [S,B,D]
// ---------------------------------------------------------------------------
__global__ __launch_bounds__(256) void transpose_out_kernel(
    const float* __restrict__ x, float* __restrict__ out) {
  size_t idx = (size_t)blockIdx.x * 256 + threadIdx.x;
  int d = (int)(idx & 255);
  size_t t = idx >> 8;
  int s = (int)(t % SQ);
  int b = (int)(t / SQ);
  out[((size_t)s * BQ + b) * DQ + d] = x[((size_t)b * SQ + s) * DQ + d];
}

// ---------------------------------------------------------------------------
extern "C" void kernel_launch(void* const* d_in, const int* in_sizes, int n_in,
                              void* d_out, int out_size, void* d_ws, size_t ws_size,
                              hipStream_t stream) {
  (void)in_sizes; (void)n_in; (void)out_size; (void)ws_size;
  const float* facts      = (const float*)d_in[0];
  const float* edge_key   = (const float*)d_in[1];
  const float* edge_value = (const float*)d_in[2];
  const unsigned char* edge_mask = (const unsigned char*)d_in[3];
  const float* Wq = (const float*)d_in[4];
  const float* Wk = (const float*)d_in[5];
  const float* Wv = (const float*)d_in[6];
  const float* Wo = (const float*)d_in[7];
  const float* bq = (const float*)d_in[8];
  const float* bk = (const float*)d_in[9];
  const float* bv = (const float*)d_in[10];
  const float* bo = (const float*)d_in[11];
  const float* ln1_g = (const float*)d_in[12];
  const float* ln1_b = (const float*)d_in[13];
  const float* W1 = (const float*)d_in[14];
  const float* b1 = (const float*)d_in[15];
  const float* W2 = (const float*)d_in[16];
  const float* b2 = (const float*)d_in[17];
  const float* ln2_g = (const float*)d_in[18];
  const float* ln2_b = (const float*)d_in[19];

  const size_t NX = (size_t)BSQ * DQ;  // 4,194,304 floats
  float* ws  = (float*)d_ws;
  float* xA  = ws;            // current x
  float* xB  = ws + NX;       // x after attention sublayer
  float* qb  = ws + 2 * NX;
  float* kb  = ws + 3 * NX;
  float* vb  = ws + 4 * NX;
  float* cb  = ws + 5 * NX;   // attention context

  const size_t GEMM_SMEM = (size_t)64 * ALD * sizeof(float);
  const size_t ATTN_SMEM = (size_t)(3 * SQ * QLD + 32 * SSTR + 32 * CLD) * sizeof(float);
  const size_t PROJ_SMEM = (size_t)(2 * 16 * ALD) * sizeof(float);
  const size_t FFN_SMEM  = (size_t)(2 * 16 * ALD + 16 * HLD) * sizeof(float);

  hipMemcpyAsync(xA, facts, NX * sizeof(float), hipMemcpyDeviceToDevice, stream);

  dim3 gproj(BSQ / 64, DQ / 64);  // 256 x 4
  for (int l = 0; l < 4; ++l) {
    const float* wq = Wq + (size_t)l * DQ * DQ;
    const float* wk = Wk + (size_t)l * DQ * DQ;
    const float* wv = Wv + (size_t)l * DQ * DQ;
    const float* wo = Wo + (size_t)l * DQ * DQ;
    const float* w1 = W1 + (size_t)l * DQ * FFNQ;
    const float* w2 = W2 + (size_t)l * FFNQ * DQ;

    gemm_kernel<<<gproj, 128, GEMM_SMEM, stream>>>(xA, wq, bq + l * DQ, qb, BSQ, DQ, DQ, 0);
    gemm_kernel<<<gproj, 128, GEMM_SMEM, stream>>>(xA, wk, bk + l * DQ, kb, BSQ, DQ, DQ, 0);
    gemm_kernel<<<gproj, 128, GEMM_SMEM, stream>>>(xA, wv, bv + l * DQ, vb, BSQ, DQ, DQ, 0);

    attn_kernel<<<BQ * HQ, 128, ATTN_SMEM, stream>>>(qb, kb, vb, edge_key, edge_value,
                                                     edge_mask, cb);

    proj_ln_kernel<<<BSQ / 16, 128, PROJ_SMEM, stream>>>(cb, wo, bo + l * DQ, xA,
                                                         ln1_g + l * DQ, ln1_b + l * DQ, xB);

    ffn_kernel<<<BSQ / 16, 128, FFN_SMEM, stream>>>(xB, w1, b1 + l * FFNQ, w2, b2 + l * DQ,
                                                    ln2_g + l * DQ, ln2_b + l * DQ, xA);
  }

  transpose_out_kernel<<<(unsigned)(NX / 256), 256, 0, stream>>>(xA, (float*)d_out);
}